// SimpleGCN_31576599560550
// MI455X (gfx1250) — compile-verified
//
#include <hip/hip_runtime.h>
#include <hip/hip_bf16.h>

typedef __attribute__((ext_vector_type(2))) float v2f;
typedef __attribute__((ext_vector_type(8))) float v8f;

#define IN_FEATS 256
#define N_HIDDEN 128
#define N_CLASSES 64

// ---------------------------------------------------------------------------
// Elementwise helpers
// ---------------------------------------------------------------------------
__global__ void gcn_zero_f32(float* __restrict__ p, int n) {
    int i = blockIdx.x * blockDim.x + threadIdx.x;
    if (i < n) p[i] = 0.0f;
}

__global__ void gcn_degrees(const int* __restrict__ src, const int* __restrict__ dst,
                            float* __restrict__ deg_out, float* __restrict__ deg_in, int nE) {
    int e = blockIdx.x * blockDim.x + threadIdx.x;
    if (e < nE) {
        atomicAdd(deg_out + src[e], 1.0f);
        atomicAdd(deg_in + dst[e], 1.0f);
    }
}

// in-place: c = rsqrt(max(deg, 1))
__global__ void gcn_norms(float* __restrict__ c, int n) {
    int i = blockIdx.x * blockDim.x + threadIdx.x;
    if (i < n) c[i] = rsqrtf(fmaxf(c[i], 1.0f));
}

// out = (RELU ? max(0, .) : .)(in * c_dst[node] + bias[f]);  may alias in==out
template <int F, bool RELU>
__global__ void gcn_scale_bias(const float* __restrict__ in, const float* __restrict__ cdst,
                               const float* __restrict__ bias, float* __restrict__ out, int n) {
    int i = blockIdx.x * blockDim.x + threadIdx.x;
    if (i < n * F) {
        int node = i / F;
        int f = i & (F - 1);
        float v = fmaf(in[i], cdst[node], bias[f]);
        if (RELU) v = fmaxf(v, 0.0f);
        out[i] = v;
    }
}

// ---------------------------------------------------------------------------
// Pack B[K,NOUT] into per-wave WMMA fragment order:
//   Bp[(ks*NT + t)*32 + lane] = { B[4*ks + 2*half + 0][t*16 + l16],
//                                 B[4*ks + 2*half + 1][t*16 + l16] }
// so the GEMM inner loop does one coalesced global_load_b64 per tile.
// ---------------------------------------------------------------------------
template <int K, int NOUT>
__global__ void gcn_pack_B(const float* __restrict__ B, float* __restrict__ Bp) {
    constexpr int NT = NOUT / 16;
    constexpr int TOTAL = (K / 4) * NT * 32;
    int idx = blockIdx.x * blockDim.x + threadIdx.x;
    if (idx >= TOTAL) return;
    int lane = idx & 31;
    int tmp = idx >> 5;
    int t = tmp % NT;
    int ks = tmp / NT;
    int half = lane >> 4;
    int l16 = lane & 15;
    int krow = ks * 4 + 2 * half;
    int col = t * 16 + l16;
    Bp[(size_t)idx * 2 + 0] = B[(size_t)krow * NOUT + col];
    Bp[(size_t)idx * 2 + 1] = B[(size_t)(krow + 1) * NOUT + col];
}

// ---------------------------------------------------------------------------
// WMMA GEMM:  C[M,NOUT] = A[M,K] @ B[K,NOUT]   (f32, V_WMMA_F32_16X16X4_F32)
// One wave computes a 16-row strip across all NOUT columns (NOUT/16 tiles).
// A fragments: global_load_b64 per k-step; B fragments: pre-packed, one
// coalesced global_load_b64 per tile per k-step (WGP$-resident).
// ---------------------------------------------------------------------------
template <int K, int NOUT>
__global__ __launch_bounds__(256) void gcn_gemm_wmma(const float* __restrict__ A,
                                                     const float* __restrict__ Bp,
                                                     float* __restrict__ C, int M) {
    constexpr int NT = NOUT / 16;
    const int wave = threadIdx.x >> 5;
    const int lane = threadIdx.x & 31;
    const int mbase = (blockIdx.x * 8 + wave) * 16;
    if (mbase >= M) return;  // wave-uniform: EXEC stays all-ones for WMMA

    const int half = lane >> 4;  // ISA 16x4 f32 A layout: lanes 16-31 hold K pair {2,3}
    const int l16 = lane & 15;

    int arow = mbase + l16;
    if (arow >= M) arow = M - 1;  // clamp loads; stores are guarded below
    const float* __restrict__ ap = A + (size_t)arow * K + 2 * half;
    const v2f* __restrict__ bpk = (const v2f*)Bp + lane;

    v8f acc[NT];
#pragma unroll
    for (int t = 0; t < NT; ++t) acc[t] = {};

    for (int k = 0; k < K; k += 4) {
        v2f a;
        a.x = ap[k];      // A[m][k + 2*half + 0]
        a.y = ap[k + 1];  // A[m][k + 2*half + 1]
        const v2f* __restrict__ bp = bpk + (size_t)(k >> 2) * NT * 32;
#pragma unroll
        for (int t = 0; t < NT; ++t) {
            v2f b = bp[t * 32];
            acc[t] = __builtin_amdgcn_wmma_f32_16x16x4_f32(
                false, a, false, b, (short)0, acc[t], false, false);
        }
    }

    // C/D layout: VGPR v -> row (v + 8*half), col = lane%16 within tile
#pragma unroll
    for (int t = 0; t < NT; ++t) {
#pragma unroll
        for (int v = 0; v < 8; ++v) {
            int row = mbase + v + 8 * half;
            if (row < M) C[(size_t)row * NOUT + t * 16 + l16] = acc[t][v];
        }
    }
}

// ---------------------------------------------------------------------------
// Edge aggregation: out[dst[e]] += H[src[e]] * c_src[src[e]]
// One wave32 per edge; float4/float2 gathers + global f32 atomic scatter.
// ---------------------------------------------------------------------------
template <int F>
__global__ __launch_bounds__(256) void gcn_aggregate(const float* __restrict__ H,
                                                     const int* __restrict__ src,
                                                     const int* __restrict__ dst,
                                                     const float* __restrict__ c_src,
                                                     float* __restrict__ out, int nE) {
    const int gid = blockIdx.x * blockDim.x + threadIdx.x;
    const int e = gid >> 5;
    const int lane = threadIdx.x & 31;
    if (e >= nE) return;
    const int s = src[e];
    const int d = dst[e];
    const float cs = c_src[s];

    if constexpr (F == 128) {
        const float4 h = *reinterpret_cast<const float4*>(H + (size_t)s * F + lane * 4);
        float* op = out + (size_t)d * F + lane * 4;
        atomicAdd(op + 0, h.x * cs);
        atomicAdd(op + 1, h.y * cs);
        atomicAdd(op + 2, h.z * cs);
        atomicAdd(op + 3, h.w * cs);
    } else {  // F == 64
        const float2 h = *reinterpret_cast<const float2*>(H + (size_t)s * F + lane * 2);
        float* op = out + (size_t)d * F + lane * 2;
        atomicAdd(op + 0, h.x * cs);
        atomicAdd(op + 1, h.y * cs);
    }
}

// ---------------------------------------------------------------------------
// Orchestration
// ---------------------------------------------------------------------------
extern "C" void kernel_launch(void* const* d_in, const int* in_sizes, int n_in,
                              void* d_out, int out_size, void* d_ws, size_t ws_size,
                              hipStream_t stream) {
    const float* features = (const float*)d_in[0];  // [N, 256]
    const float* W1 = (const float*)d_in[1];        // [256, 128]
    const float* b1 = (const float*)d_in[2];        // [128]
    const float* W2 = (const float*)d_in[3];        // [128, 64]
    const float* b2 = (const float*)d_in[4];        // [64]
    const int* src = (const int*)d_in[5];           // [E]
    const int* dst = (const int*)d_in[6];           // [E]
    float* out = (float*)d_out;                     // [N, 64]

    const int N = in_sizes[0] / IN_FEATS;
    const int E = in_sizes[5];

    // Workspace (floats): c_src[N] | c_dst[N] | bufA[N*128] | bufB[N*128] | Bp1 | Bp2
    float* ws = (float*)d_ws;
    float* c_src = ws;
    float* c_dst = ws + N;
    float* bufA = ws + 2 * (size_t)N;                    // hW1, then h_act (relu'd)
    float* bufB = ws + 2 * (size_t)N + (size_t)N * 128;  // agg1, then hW2
    float* Bp1 = ws + 2 * (size_t)N + 2 * (size_t)N * 128;              // 32768 f
    float* Bp2 = Bp1 + (IN_FEATS / 4) * (N_HIDDEN / 16) * 64;           // 8192 f

    const int T = 256;
    auto blocks = [](long n, int t) { return (unsigned)((n + t - 1) / t); };

    // 1) zero accumulators (degrees, agg1, d_out)
    gcn_zero_f32<<<blocks(2L * N, T), T, 0, stream>>>(c_src, 2 * N);
    gcn_zero_f32<<<blocks((long)N * 128, T), T, 0, stream>>>(bufB, N * 128);
    gcn_zero_f32<<<blocks((long)N * 64, T), T, 0, stream>>>(out, N * 64);

    // 2) degrees -> norms
    gcn_degrees<<<blocks(E, T), T, 0, stream>>>(src, dst, c_src, c_dst, E);
    gcn_norms<<<blocks(2L * N, T), T, 0, stream>>>(c_src, 2 * N);

    // 3) pack weights into WMMA fragment order
    gcn_pack_B<IN_FEATS, N_HIDDEN><<<blocks(16384, T), T, 0, stream>>>(W1, Bp1);
    gcn_pack_B<N_HIDDEN, N_CLASSES><<<blocks(4096, T), T, 0, stream>>>(W2, Bp2);

    // 4) layer 1: hW1 = X @ W1  (WMMA f32)
    gcn_gemm_wmma<IN_FEATS, N_HIDDEN><<<blocks(N, 128), T, 0, stream>>>(features, Bp1, bufA, N);

    // 5) agg1[dst] += hW1[src] * c_src[src]
    gcn_aggregate<N_HIDDEN><<<blocks((long)E * 32, T), T, 0, stream>>>(bufA, src, dst, c_src, bufB, E);

    // 6) h_act = relu(agg1 * c_dst + b1)   (into bufA)
    gcn_scale_bias<N_HIDDEN, true><<<blocks((long)N * 128, T), T, 0, stream>>>(bufB, c_dst, b1, bufA, N);

    // 7) layer 2: hW2 = h_act @ W2  (WMMA f32, into bufB)
    gcn_gemm_wmma<N_HIDDEN, N_CLASSES><<<blocks(N, 128), T, 0, stream>>>(bufA, Bp2, bufB, N);

    // 8) out[dst] += hW2[src] * c_src[src]   (out pre-zeroed)
    gcn_aggregate<N_CLASSES><<<blocks((long)E * 32, T), T, 0, stream>>>(bufB, src, dst, c_src, out, E);

    // 9) out = out * c_dst + b2  (in place)
    gcn_scale_bias<N_CLASSES, false><<<blocks((long)N * 64, T), T, 0, stream>>>(out, c_dst, b2, out, N);
}